// GlobalModel_11433202942743
// MI455X (gfx1250) — compile-verified
//
#include <hip/hip_runtime.h>
#include <hip/hip_bf16.h>
#include <math.h>

typedef float v2f __attribute__((ext_vector_type(2)));
typedef float v8f __attribute__((ext_vector_type(8)));

#define EPS 1e-12f
#define NEIGH_WEIGHT 0.2f

// ---------------- zero workspace ----------------
__global__ void zero_kernel(float* p, size_t n) {
    size_t i = (size_t)blockIdx.x * blockDim.x + threadIdx.x;
    size_t stride = (size_t)gridDim.x * blockDim.x;
    for (; i < n; i += stride) p[i] = 0.0f;
}

// ---------------- stats: sum / sumsq of pos_diff[nor_idx] ----------------
__global__ void stats_kernel(const float* __restrict__ pos_diff,
                             const int* __restrict__ nor_idx, int M,
                             float* __restrict__ scratch) {
    int t = blockIdx.x * blockDim.x + threadIdx.x;
    float v = 0.0f;
    if (t < M) v = pos_diff[nor_idx[t]];
    float s1 = v, s2 = v * v;
    #pragma unroll
    for (int off = 16; off > 0; off >>= 1) {
        s1 += __shfl_xor(s1, off, 32);
        s2 += __shfl_xor(s2, off, 32);
    }
    if ((threadIdx.x & 31) == 0) {
        __hip_atomic_fetch_add(&scratch[0], s1, __ATOMIC_RELAXED, __HIP_MEMORY_SCOPE_AGENT);
        __hip_atomic_fetch_add(&scratch[1], s2, __ATOMIC_RELAXED, __HIP_MEMORY_SCOPE_AGENT);
    }
}

// ---------------- fused WMMA GEMM + bias + relu + row L2 normalize ----------------
// Block: 256 threads = 8 waves. Block handles 16 rows x 128 cols; wave w -> cols [16w,16w+16).
__global__ __launch_bounds__(256) void gemm_h_kernel(
    const float* __restrict__ feats, const float* __restrict__ W,
    const float* __restrict__ bias, float* __restrict__ h, int N) {
    __shared__ float As[16 * 260];   // A strip 16x256, stride 260 (bank-conflict free)
    __shared__ float T[16 * 132];    // output tile 16x128, stride 132
    __shared__ float P[16 * 17];     // per-row partial sq-sums
    __shared__ float RowInv[16];

    const int tid  = threadIdx.x;
    const int wave = tid >> 5;
    const int lane = tid & 31;
    const int row0 = blockIdx.x * 16;
    const int cb   = wave * 16;
    const int hl   = lane & 15;
    const int hi   = lane >> 4;   // 0 or 1 (half-wave)

    // stage A strip in LDS (coalesced float4 loads)
    #pragma unroll
    for (int j = 0; j < 4; ++j) {
        int idx = tid * 4 + j * 1024;          // 0..4095
        int r = idx >> 8;
        int c = idx & 255;
        int grow = row0 + r;
        float4 v = make_float4(0.f, 0.f, 0.f, 0.f);
        if (grow < N) v = *(const float4*)(feats + (size_t)grow * 256 + c);
        float* dstp = &As[r * 260 + c];
        dstp[0] = v.x; dstp[1] = v.y; dstp[2] = v.z; dstp[3] = v.w;
    }
    __syncthreads();

    v8f acc = {};
    for (int k = 0; k < 256; k += 4) {
        int ka = k + hi * 2;
        v2f a, b;
        a.x = As[hl * 260 + ka];
        a.y = As[hl * 260 + ka + 1];
        b.x = W[(size_t)ka * 128 + cb + hl];
        b.y = W[(size_t)(ka + 1) * 128 + cb + hl];
        acc = __builtin_amdgcn_wmma_f32_16x16x4_f32(false, a, false, b,
                                                    (short)0, acc, false, false);
    }

    // bias + relu, stash tile in LDS
    float bv = bias[cb + hl];
    int rbase = hi * 8;
    #pragma unroll
    for (int r = 0; r < 8; ++r) {
        float v = acc[r] + bv;
        v = v > 0.0f ? v : 0.0f;
        T[(rbase + r) * 132 + cb + hl] = v;
    }
    __syncthreads();

    // per-row sum of squares (16 threads per row, 8 cols each)
    {
        int row = tid >> 4;
        int c0  = (tid & 15) * 8;
        float s = 0.0f;
        #pragma unroll
        for (int j = 0; j < 8; ++j) { float v = T[row * 132 + c0 + j]; s += v * v; }
        P[row * 17 + (tid & 15)] = s;
    }
    __syncthreads();
    if (tid < 16) {
        float s = 0.0f;
        #pragma unroll
        for (int j = 0; j < 16; ++j) s += P[tid * 17 + j];
        RowInv[tid] = 1.0f / fmaxf(sqrtf(s), EPS);
    }
    __syncthreads();

    // normalized store
    {
        int row  = tid >> 4;
        int grow = row0 + row;
        if (grow < N) {
            int c0 = (tid & 15) * 8;
            float inv = RowInv[row];
            float* op = h + (size_t)grow * 128 + c0;
            #pragma unroll
            for (int j = 0; j < 8; ++j) op[j] = T[row * 132 + c0 + j] * inv;
        }
    }
}

// ---------------- edge aggregation: accum[dst] += h[src], deg[dst] += 1 ----------------
// one wave per edge; lane covers 4 columns -> 128B coalesced gather, f32 atomics scatter
__global__ __launch_bounds__(256) void aggregate_kernel(
    const float* __restrict__ h, const int* __restrict__ src,
    const int* __restrict__ dst, float* __restrict__ accum,
    float* __restrict__ deg, int E) {
    int gw = (int)(((size_t)blockIdx.x * blockDim.x + threadIdx.x) >> 5);
    int lane = threadIdx.x & 31;
    if (gw >= E) return;
    int s = src[gw], d = dst[gw];
    const float* hp = h + (size_t)s * 128;
    float* ap = accum + (size_t)d * 128;
    #pragma unroll
    for (int j = 0; j < 4; ++j) {
        int c = lane + 32 * j;
        __hip_atomic_fetch_add(&ap[c], hp[c], __ATOMIC_RELAXED, __HIP_MEMORY_SCOPE_AGENT);
    }
    if (lane == 0)
        __hip_atomic_fetch_add(&deg[d], 1.0f, __ATOMIC_RELAXED, __HIP_MEMORY_SCOPE_AGENT);
}

// ---------------- per-node finalize: mean, attention blend, score, loss partials ----------------
__global__ __launch_bounds__(256) void finalize_kernel(
    const float* __restrict__ h, const float* __restrict__ accum,
    const float* __restrict__ deg, const float* __restrict__ center,
    const float* __restrict__ pos_diff, const int* __restrict__ out_nodes,
    const int* __restrict__ labels, const int* __restrict__ epoch,
    float* __restrict__ scratch, float* __restrict__ scores_out,
    int N, int Mnor) {
    int gw = (int)(((size_t)blockIdx.x * blockDim.x + threadIdx.x) >> 5);
    int lane = threadIdx.x & 31;
    if (gw >= N) return;
    int i = gw;

    float dg = deg[i];
    float invd = dg > 0.0f ? 1.0f / fmaxf(dg, 1.0f) : 0.0f;

    const float* hp = h + (size_t)i * 128;
    const float* ap = accum + (size_t)i * 128;
    float hv[4], mv[4];
    float dot = 0.0f;
    #pragma unroll
    for (int j = 0; j < 4; ++j) {
        int c = lane + 32 * j;
        hv[j] = hp[c];
        mv[j] = ap[c] * invd;
        dot += hv[j] * mv[j];
    }
    #pragma unroll
    for (int off = 16; off > 0; off >>= 1) dot += __shfl_xor(dot, off, 32);

    // pre-attention from stats
    int on = out_nodes[i];
    float cnt = (float)Mnor;
    float mu  = scratch[0] / cnt;
    float var = (scratch[1] - cnt * mu * mu) / (cnt - 1.0f);
    float sd  = sqrtf(var);
    float z   = (pos_diff[on] - mu) / sd;
    float pre = 1.0f - 1.0f / (1.0f + expf(-z));

    float beta = powf(0.9f, (float)epoch[0]);
    if (beta < 0.1f) beta = 0.0f;
    float attn = beta * pre + (1.0f - beta) * dot;
    float nei  = attn * NEIGH_WEIGHT;

    float score = 0.0f;
    #pragma unroll
    for (int j = 0; j < 4; ++j) {
        int c = lane + 32 * j;
        float h2 = nei * mv[j] + (1.0f - nei) * hv[j];
        score += h2 * center[c];
    }
    #pragma unroll
    for (int off = 16; off > 0; off >>= 1) score += __shfl_xor(score, off, 32);

    if (lane == 0) {
        scores_out[i] = score;
        int lab = labels[on];
        float spl = log1pf(expf(-fabsf(score)));   // stable log(1+e^-|x|)
        if (lab == 1) {
            // target 0 -> softplus(score)
            float sp = fmaxf(score, 0.0f) + spl;
            __hip_atomic_fetch_add(&scratch[4], sp, __ATOMIC_RELAXED, __HIP_MEMORY_SCOPE_AGENT);
            __hip_atomic_fetch_add(&scratch[5], 1.0f, __ATOMIC_RELAXED, __HIP_MEMORY_SCOPE_AGENT);
        } else {
            // target 1 -> softplus(-score)
            float sp = fmaxf(-score, 0.0f) + spl;
            __hip_atomic_fetch_add(&scratch[2], sp, __ATOMIC_RELAXED, __HIP_MEMORY_SCOPE_AGENT);
            __hip_atomic_fetch_add(&scratch[3], 1.0f, __ATOMIC_RELAXED, __HIP_MEMORY_SCOPE_AGENT);
        }
    }
}

__global__ void loss_kernel(const float* __restrict__ scratch, float* __restrict__ out0) {
    if (threadIdx.x == 0 && blockIdx.x == 0) {
        out0[0] = scratch[2] / fmaxf(scratch[3], 1.0f) +
                  scratch[4] / fmaxf(scratch[5], 1.0f);
    }
}

extern "C" void kernel_launch(void* const* d_in, const int* in_sizes, int n_in,
                              void* d_out, int out_size, void* d_ws, size_t ws_size,
                              hipStream_t stream) {
    const float* feats    = (const float*)d_in[0];
    const float* W        = (const float*)d_in[1];
    const float* b        = (const float*)d_in[2];
    const float* center   = (const float*)d_in[3];
    const float* pos_diff = (const float*)d_in[4];
    const int*   src      = (const int*)d_in[5];
    const int*   dst      = (const int*)d_in[6];
    const int*   out_nodes= (const int*)d_in[7];
    const int*   nor_idx  = (const int*)d_in[8];
    const int*   labels   = (const int*)d_in[9];
    const int*   epoch    = (const int*)d_in[10];

    const int N = in_sizes[0] / 256;   // 100000
    const int E = in_sizes[5];         // 1600000
    const int M = in_sizes[8];         // 50000

    float* wsf     = (float*)d_ws;
    float* scratch = wsf;                          // 64 floats
    float* h       = wsf + 64;                     // N*128
    float* accum   = h + (size_t)N * 128;          // N*128
    float* deg     = accum + (size_t)N * 128;      // N

    float* out     = (float*)d_out;                // [0]=loss, [1..N]=scores

    // 1) zero scratch + accum + deg (h fully overwritten by GEMM)
    size_t zn = 64 + 2 * (size_t)N * 128 + (size_t)N;
    zero_kernel<<<2048, 256, 0, stream>>>(wsf, zn);

    // 2) stats over pos_diff[nor_idx]
    stats_kernel<<<(M + 255) / 256, 256, 0, stream>>>(pos_diff, nor_idx, M, scratch);

    // 3) h = l2norm(relu(feats @ W + b))  -- WMMA
    gemm_h_kernel<<<(N + 15) / 16, 256, 0, stream>>>(feats, W, b, h, N);

    // 4) neighbor aggregation (one wave per edge)
    aggregate_kernel<<<(E + 7) / 8, 256, 0, stream>>>(h, src, dst, accum, deg, E);

    // 5) per-node blend + scores + loss partials (one wave per node)
    finalize_kernel<<<(N + 7) / 8, 256, 0, stream>>>(h, accum, deg, center, pos_diff,
                                                     out_nodes, labels, epoch, scratch,
                                                     out + 1, N, M);

    // 6) final scalar loss
    loss_kernel<<<1, 64, 0, stream>>>(scratch, out);
}